// EdgeWeightLearner_55422257987594
// MI455X (gfx1250) — compile-verified
//
#include <hip/hip_runtime.h>
#include <hip/hip_bf16.h>
#include <math.h>

// ---------------------------------------------------------------------------
// EdgeWeightLearner for MI455X (gfx1250, wave32)
//
// reference:  s = x[row]·w_row + x[col]·w_col ; w = sigmoid(s)
//             out[e] = w[e] * w[lri[e]]
//
// Memory-bound: ~100 MB total -> ~4.3us @ 23.3 TB/s. Plan:
//   K1: node projection S[n] = (x[n]·w_row, x[n]·w_col) with
//       V_WMMA_F32_16X16X4_F32. Each wave owns one 16-node tile:
//         - stage 16x128 f32 tile via 16 fully-coalesced b128 row loads
//           into LDS (row stride padded to 132 floats -> conflict-free
//           ds_load_b64 for the A operand: bank = (4*n16 + k0) % 64).
//         - 32 WMMA steps over K=128, branch-free hot loop (boundary rows
//           are clamped, B lanes >=2 zeroed via selects, EXEC stays full).
//   K2: edge score from L2-resident 800KB S table, sigmoid -> w (ws).
//   K3: out[e] = w[e] * w[lri[e]] (w table 6.4MB, L2-resident).
// ---------------------------------------------------------------------------

typedef float v2f __attribute__((ext_vector_type(2)));
typedef float v8f __attribute__((ext_vector_type(8)));

#define CDIM    128
#define LDSPITCH 132   // 128 + 4 pad floats: conflict-free A reads

// ---- Kernel 1: node projection via f32 WMMA -------------------------------
// s[2*n+0] = dot(x[n], W[0:128]),  s[2*n+1] = dot(x[n], W[128:256])
__global__ __launch_bounds__(256) void ewl_node_proj(const float* __restrict__ x,
                                                     const float* __restrict__ W,
                                                     float* __restrict__ s,
                                                     int nTiles, int N) {
    __shared__ float lds_all[8][16 * LDSPITCH];    // 8 waves x 8448B = 67.6KB
    const int wave = threadIdx.x >> 5;
    const int lane = threadIdx.x & 31;
    const int tile = blockIdx.x * 8 + wave;
    if (tile >= nTiles) return;                    // wave-uniform exit

    float* __restrict__ lds = &lds_all[wave][0];

    // ---- stage the 16x128 tile: 16 coalesced b128 row loads (8KB stream) --
    #pragma unroll
    for (int r = 0; r < 16; ++r) {
        long row = (long)tile * 16 + r;
        if (row >= (long)N) row = (long)N - 1;     // clamp: dup row, not stored
        const float4 v = ((const float4*)(x + row * CDIM))[lane];
        ((float4*)(lds + r * LDSPITCH))[lane] = v; // 528B pitch, 16B aligned
    }

    const int half = lane >> 4;                    // K sub-pair select
    const int n16  = lane & 15;                    // A: row M ; B: col N
    const float* __restrict__ arow = lds + n16 * LDSPITCH;
    // B column n16: col0 <- w_row = W[0..127], col1 <- w_col = W[128..255].
    // Clamped pointer keeps every lane's load in-bounds; lanes >=2 zeroed.
    const float* __restrict__ wsrc = W + (n16 < 2 ? n16 : 1) * CDIM;
    const bool bon = (n16 < 2);

    v8f acc = {};
    #pragma unroll
    for (int kb = 0; kb < CDIM; kb += 4) {
        const int k0 = kb + 2 * half;              // this lane holds K = k0,k0+1
        v2f a;                                     // ds_load_b64, conflict-free
        a.x = arow[k0];
        a.y = arow[k0 + 1];
        const float bx = wsrc[k0];                 // L0-resident 1KB W
        const float by = wsrc[k0 + 1];
        v2f b;                                     // v_cndmask, no EXEC games
        b.x = bon ? bx : 0.0f;
        b.y = bon ? by : 0.0f;
        acc = __builtin_amdgcn_wmma_f32_16x16x4_f32(
            /*neg_a=*/false, a, /*neg_b=*/false, b,
            /*c_mod=*/(short)0, acc, /*reuse_a=*/false, /*reuse_b=*/false);
    }

    // C/D layout: VGPR v, lane half h -> M = v + 8*h, N = lane&15.
    if (n16 < 2) {
        #pragma unroll
        for (int v = 0; v < 8; ++v) {
            const long node = (long)tile * 16 + v + 8 * half;
            if (node < (long)N) s[node * 2 + n16] = acc[v];
        }
    }
}

// ---- Kernel 2: per-edge score + sigmoid -----------------------------------
__global__ __launch_bounds__(256) void ewl_edge_sigmoid(const long long* __restrict__ ei,
                                                        const float* __restrict__ s,
                                                        float* __restrict__ w,
                                                        long E) {
    const long e = (long)blockIdx.x * blockDim.x + threadIdx.x;
    if (e >= E) return;
    const long r = ei[e];          // edge_index[0][e]
    const long c = ei[E + e];      // edge_index[1][e]
    const float sc = s[2 * r] + s[2 * c + 1];      // gathers hit L2 (800KB table)
    w[e] = 1.0f / (1.0f + expf(-sc));              // sigmoid, full-precision exp
}

// ---- Kernel 3: pairwise product via left_right gather ---------------------
__global__ __launch_bounds__(256) void ewl_edge_mul(const float* __restrict__ w,
                                                    const long long* __restrict__ lri,
                                                    float* __restrict__ out,
                                                    long E) {
    const long e = (long)blockIdx.x * blockDim.x + threadIdx.x;
    if (e >= E) return;
    out[e] = w[e] * w[lri[e]];                     // gather hits L2 (6.4MB table)
}

// ---------------------------------------------------------------------------
extern "C" void kernel_launch(void* const* d_in, const int* in_sizes, int n_in,
                              void* d_out, int out_size, void* d_ws, size_t ws_size,
                              hipStream_t stream) {
    const float*     x   = (const float*)d_in[0];        // (N, 128) f32
    const long long* ei  = (const long long*)d_in[1];    // (2, E)   i64
    const long long* lri = (const long long*)d_in[2];    // (E,)     i64
    const float*     W   = (const float*)d_in[3];        // (1, 256) f32

    const int  N = in_sizes[0] / CDIM;
    const long E = (long)in_sizes[2];
    float* out = (float*)d_out;

    // workspace layout: [0, E) edge weights ; [E, E+2N) node projections
    float* w = (float*)d_ws;
    float* s = w + E;

    const int nTiles = (N + 15) / 16;
    ewl_node_proj<<<dim3((nTiles + 7) / 8), dim3(256), 0, stream>>>(x, W, s, nTiles, N);

    const int eBlocks = (int)((E + 255) / 256);
    ewl_edge_sigmoid<<<dim3(eBlocks), dim3(256), 0, stream>>>(ei, s, w, E);
    ewl_edge_mul<<<dim3(eBlocks), dim3(256), 0, stream>>>(w, lri, out, E);
}